// PatchedInternLM2AttentionMaca_4466765988391
// MI455X (gfx1250) — compile-verified
//
#include <hip/hip_runtime.h>
#include <math.h>

// ---- InternLM2-7B decode-step dims ----
#define B_      32
#define DMODEL  4096
#define H_      32
#define KV_     8
#define G_      4            // H_/KV_ (GQA group)
#define HD_     128
#define BLOCK_  64
#define NBLK_   64
#define L_      (NBLK_*BLOCK_)          // 4096 keys per sequence
#define QKVN    ((H_ + 2*KV_) * HD_)    // 6144 fused-qkv cols
#define THETA_  1.0e6f

typedef float v2f __attribute__((ext_vector_type(2)));
typedef float v8f __attribute__((ext_vector_type(8)));

// =====================================================================
// fp32 GEMM via V_WMMA_F32_16X16X4_F32.  C[M,N] = A[M,K] * B[K,N].
// N, K are template constants so all B-fragment addresses reduce to
// base + immediate-offset global loads (24-bit ioffset) instead of
// per-load v_mul_u64 address math.  One wave owns one 16x16 C tile;
// 4 waves per block tile along N.  Two independent accumulator chains
// (k and k+4) break the serial WMMA RAW dependency; one near-cache
// prefetch per 64-k outer step.
// Fragment layouts per CDNA5 ISA 7.12.2:
//   A 16x4 : lane(0..15)=M, vgpr i => K = 2*(lane>>4)+i
//   B 4x16 : lane&15 = N,   vgpr i => K = 2*(lane>>4)+i
//   C 16x16: vgpr r, lanes 0-15 -> M=r, lanes 16-31 -> M=r+8, N=lane&15
// =====================================================================
template<int N, int K>
__global__ __launch_bounds__(128)
void wmma_gemm_f32(const float* __restrict__ A, const float* __restrict__ Bm,
                   float* __restrict__ C)
{
  const int lane = threadIdx.x & 31;
  const int wave = threadIdx.x >> 5;
  const int n0 = (blockIdx.x * 4 + wave) * 16;
  const int m0 = blockIdx.y * 16;
  const int r  = lane & 15;            // A row in tile / B,C col in tile
  const int kg = (lane >> 4) * 2;      // lane-group K offset

  v8f acc0 = {}, acc1 = {};
  // fold the lane-group offset into the base pointers once
  const float* aq = A + (size_t)(m0 + r) * K + kg;
  const float* bq = Bm + (size_t)kg * N + n0 + r;

  for (int kk = 0; kk < K; kk += 64) {
    if (kk + 64 < K)                   // one near-cache prefetch per 64 k
      __builtin_prefetch(bq + (size_t)(kk + 64) * N, 0, 3);
    #pragma unroll
    for (int u = 0; u < 64; u += 8) {
      v2f a0, b0, a1, b1;
      a0[0] = aq[kk + u + 0];
      a0[1] = aq[kk + u + 1];
      b0[0] = bq[(size_t)(kk + u + 0) * N];
      b0[1] = bq[(size_t)(kk + u + 1) * N];
      a1[0] = aq[kk + u + 4];
      a1[1] = aq[kk + u + 5];
      b1[0] = bq[(size_t)(kk + u + 4) * N];
      b1[1] = bq[(size_t)(kk + u + 5) * N];
      acc0 = __builtin_amdgcn_wmma_f32_16x16x4_f32(false, a0, false, b0,
                                                   (short)0, acc0, false, false);
      acc1 = __builtin_amdgcn_wmma_f32_16x16x4_f32(false, a1, false, b1,
                                                   (short)0, acc1, false, false);
    }
  }
  const v8f acc = acc0 + acc1;
  #pragma unroll
  for (int i = 0; i < 8; ++i) {
    const int row = i + ((lane >> 4) << 3);
    C[(size_t)(m0 + row) * N + n0 + r] = acc[i];
  }
}

// =====================================================================
// Rotate-half RoPE applied in-place to q and k rows of the qkv workspace.
// grid: B*(H+KV) blocks, HD/2 threads.
// =====================================================================
__global__ void rope_kernel(float* __restrict__ qkv, const int* __restrict__ pos_ids)
{
  const int bh = blockIdx.x;
  const int b = bh / (H_ + KV_);
  const int h = bh % (H_ + KV_);
  const int i = threadIdx.x;                 // 0..63
  const float pos = (float)pos_ids[b];
  const float inv_freq = expf(-((float)i / (float)(HD_/2)) * logf(THETA_));
  const float ang = pos * inv_freq;
  const float c = cosf(ang), s = sinf(ang);
  const int col = (h < H_) ? h * HD_ : H_ * HD_ + (h - H_) * HD_;
  float* p = qkv + (size_t)b * QKVN + col;
  const float x1 = p[i], x2 = p[i + HD_/2];
  p[i]           = x1 * c - x2 * s;
  p[i + HD_/2]   = x2 * c + x1 * s;
}

// =====================================================================
// Flash-decode paged attention.  grid = (KV, B), block = 256 (8 waves).
// Each wave streams 512 keys with online softmax; the freshly-projected
// k/v (RoPE'd, in workspace) substitutes for cache slot l == seqlen-1,
// so the input cache is never mutated.
// =====================================================================
__global__ __launch_bounds__(256)
void attn_kernel(const float* __restrict__ k_cache, const float* __restrict__ v_cache,
                 const float* __restrict__ qkv, const int* __restrict__ block_offsets,
                 const int* __restrict__ kv_seqlens, float* __restrict__ attn_out)
{
  const int kvh  = blockIdx.x;          // 0..7
  const int b    = blockIdx.y;          // 0..31
  const int tid  = threadIdx.x;
  const int wave = tid >> 5;
  const int lane = tid & 31;

  __shared__ __align__(16) float q_s[G_][HD_];          // 2 KB
  __shared__ float m_s[8][G_];
  __shared__ float s_s[8][G_];
  __shared__ __align__(16) float a_s[8][G_][HD_];       // 16 KB

  const float scale = 0.08838834764831844f;             // 1/sqrt(128)
  for (int e = tid; e < G_*HD_; e += 256) {
    const int g = e >> 7, d = e & (HD_-1);
    q_s[g][d] = qkv[(size_t)b*QKVN + (size_t)(kvh*G_ + g)*HD_ + d] * scale;
  }
  __syncthreads();

  const int seqlen = kv_seqlens[b];
  const int last   = seqlen - 1;
  const float* knew = qkv + (size_t)b*QKVN + (size_t)H_*HD_ + (size_t)kvh*HD_;
  const float* vnew = qkv + (size_t)b*QKVN + (size_t)(H_+KV_)*HD_ + (size_t)kvh*HD_;

  float m[G_], ssum[G_], ax[G_], ay[G_], az[G_], aw[G_];
  #pragma unroll
  for (int g = 0; g < G_; ++g) {
    m[g] = -1e30f; ssum[g] = 0.f;
    ax[g] = ay[g] = az[g] = aw[g] = 0.f;
  }

  const int kpw  = L_ / 8;              // 512 keys per wave
  const int base = wave * kpw;

  for (int t = 0; t < kpw; t += 32) {
    const int l0   = base + t;
    const int lme  = l0 + lane;
    // 32-key tile never crosses a 64-entry page: one scalar block lookup
    const int phys = block_offsets[b*NBLK_ + (l0 >> 6)];
    const size_t pagebase = ((size_t)phys*BLOCK_)*KV_*HD_;
    const float* krow = (lme == last) ? knew :
        k_cache + pagebase + ((size_t)(lme & (BLOCK_-1))*KV_ + kvh)*HD_;

    // prefetch V of this tile (consumed after the score pass) and K of next
    __builtin_prefetch(
        v_cache + pagebase + ((size_t)(lme & (BLOCK_-1))*KV_ + kvh)*HD_ + lane*4, 0, 3);
    if (t + 32 < kpw) {
      const int ln    = lme + 32;
      const int physn = block_offsets[b*NBLK_ + ((l0 + 32) >> 6)];
      __builtin_prefetch(
        k_cache + ((((size_t)physn*BLOCK_) + (size_t)(ln & (BLOCK_-1)))*KV_ + kvh)*HD_, 0, 3);
    }

    // ---- scores: each lane dots its key against the 4 shared q rows ----
    float sc[G_] = {0.f, 0.f, 0.f, 0.f};
    #pragma unroll 8
    for (int c = 0; c < HD_; c += 4) {
      const float4 kc = *(const float4*)(krow + c);
      #pragma unroll
      for (int g = 0; g < G_; ++g) {
        const float4 qc = *(const float4*)(&q_s[g][c]);   // LDS broadcast
        sc[g] += kc.x*qc.x + kc.y*qc.y + kc.z*qc.z + kc.w*qc.w;
      }
    }

    // ---- online softmax update (wave-uniform running stats) ----
    const bool valid = (lme < seqlen);
    float p[G_];
    #pragma unroll
    for (int g = 0; g < G_; ++g) {
      const float s0 = valid ? sc[g] : -1e30f;
      float tmax = s0;
      for (int off = 16; off > 0; off >>= 1)
        tmax = fmaxf(tmax, __shfl_xor(tmax, off, 32));
      const float newM  = fmaxf(m[g], tmax);
      const float alpha = __expf(m[g] - newM);
      const float pv    = valid ? __expf(s0 - newM) : 0.0f;
      float tsum = pv;
      for (int off = 16; off > 0; off >>= 1)
        tsum += __shfl_xor(tsum, off, 32);
      ssum[g] = ssum[g]*alpha + tsum;
      m[g]    = newM;
      ax[g] *= alpha; ay[g] *= alpha; az[g] *= alpha; aw[g] *= alpha;
      p[g] = pv;
    }

    // ---- PV: lane owns dims [lane*4, lane*4+4); V rows fully coalesced ----
    #pragma unroll 4
    for (int j = 0; j < 32; ++j) {
      const int lj = l0 + j;
      const float* vrow = (lj == last) ? vnew :
          v_cache + pagebase + ((size_t)(lj & (BLOCK_-1))*KV_ + kvh)*HD_;
      const float4 vj = *(const float4*)(vrow + lane*4);
      #pragma unroll
      for (int g = 0; g < G_; ++g) {
        const float pj = __shfl(p[g], j, 32);
        ax[g] += pj*vj.x; ay[g] += pj*vj.y; az[g] += pj*vj.z; aw[g] += pj*vj.w;
      }
    }
  }

  // ---- cross-wave combine via LDS ----
  if (lane == 0) {
    #pragma unroll
    for (int g = 0; g < G_; ++g) { m_s[wave][g] = m[g]; s_s[wave][g] = ssum[g]; }
  }
  #pragma unroll
  for (int g = 0; g < G_; ++g) {
    float4 v4; v4.x = ax[g]; v4.y = ay[g]; v4.z = az[g]; v4.w = aw[g];
    *(float4*)(&a_s[wave][g][lane*4]) = v4;
  }
  __syncthreads();

  #pragma unroll
  for (int rr = 0; rr < 2; ++rr) {
    const int e = tid + rr*256;           // 512 outputs / 256 threads
    const int g = e >> 7, d = e & (HD_-1);
    float M = -1e30f;
    for (int w = 0; w < 8; ++w) M = fmaxf(M, m_s[w][g]);
    float S = 0.f, Acc = 0.f;
    for (int w = 0; w < 8; ++w) {
      const float f = __expf(m_s[w][g] - M);
      S   += f * s_s[w][g];
      Acc += f * a_s[w][g][d];
    }
    attn_out[(size_t)b*(H_*HD_) + (size_t)(kvh*G_ + g)*HD_ + d] = Acc / S;
  }
}

// =====================================================================
extern "C" void kernel_launch(void* const* d_in, const int* in_sizes, int n_in,
                              void* d_out, int out_size, void* d_ws, size_t ws_size,
                              hipStream_t stream) {
  const float* hs   = (const float*)d_in[0];   // [B, DMODEL]
  const float* wqkv = (const float*)d_in[1];   // [DMODEL, QKVN]
  const float* wo   = (const float*)d_in[2];   // [H*HD, DMODEL]
  const float* kc   = (const float*)d_in[3];   // [NUM_BLOCKS, BLOCK, KV, HD]
  const float* vc   = (const float*)d_in[4];
  const int*   pos  = (const int*)d_in[5];     // [B]
  const int*   boff = (const int*)d_in[6];     // [B, NBLK]
  const int*   slen = (const int*)d_in[7];     // [B]
  float* out = (float*)d_out;                  // [B, DMODEL]

  float* qkv      = (float*)d_ws;              // [B, QKVN]
  float* attn_out = qkv + (size_t)B_*QKVN;     // [B, H*HD]

  // 1) fused QKV projection (WMMA f32):  [32,6144] = [32,4096] x [4096,6144]
  wmma_gemm_f32<QKVN, DMODEL><<<dim3(QKVN/64, B_/16), 128, 0, stream>>>(hs, wqkv, qkv);
  // 2) RoPE on q and k at position kv_len-1
  rope_kernel<<<B_*(H_+KV_), HD_/2, 0, stream>>>(qkv, pos);
  // 3) paged flash-decode attention (fresh k/v substituted at l == seqlen-1)
  attn_kernel<<<dim3(KV_, B_), 256, 0, stream>>>(kc, vc, qkv, boff, slen, attn_out);
  // 4) output projection (WMMA f32):  [32,4096] = [32,4096] x [4096,4096]
  wmma_gemm_f32<DMODEL, DMODEL><<<dim3(DMODEL/64, B_/16), 128, 0, stream>>>(attn_out, wo, out);
}